// Ultra_12438225289968
// MI455X (gfx1250) — compile-verified
//
#include <hip/hip_runtime.h>
#include <hip/hip_bf16.h>

#define BB 4
#define KK 16
#define NN 50000
#define EE 300000
#define RR 64
#define ERN 2048
#define RTN 4
#define DD 64
#define NLAY 4

typedef __attribute__((ext_vector_type(2))) float v2f;
typedef __attribute__((ext_vector_type(8))) float v8f;

// ---------------- utility kernels ----------------

__global__ void zero_f4(float4* p, int n4) {
    for (int i = blockIdx.x * blockDim.x + threadIdx.x; i < n4;
         i += gridDim.x * blockDim.x)
        p[i] = make_float4(0.f, 0.f, 0.f, 0.f);
}

__global__ void copy_f4(float4* d, const float4* s, int n4) {
    for (int i = blockIdx.x * blockDim.x + threadIdx.x; i < n4;
         i += gridDim.x * blockDim.x)
        d[i] = s[i];
}

// boundary_r[b, r0[b], :] = 1   (1 block x 256 threads)
__global__ void set_rel_boundary(float* bnd, const int* triples) {
    int tid = threadIdx.x;
    int b = tid >> 6, d = tid & 63;
    int r0 = triples[b * KK * 3 + 2];
    bnd[(b * RR + r0) * DD + d] = 1.f;
}

// query[b,:] = relation_repr[b, r0[b], :]
__global__ void extract_query(const float* relr, const int* triples, float* query) {
    int tid = threadIdx.x;
    int b = tid >> 6, d = tid & 63;
    int r0 = triples[b * KK * 3 + 2];
    query[b * DD + d] = relr[(b * RR + r0) * DD + d];
}

// boundary[b, h0[b], :] = query[b,:]
__global__ void set_ent_boundary(float* bnd, const float* query, const int* triples) {
    int tid = threadIdx.x;
    int b = tid >> 6, d = tid & 63;
    int h0 = triples[b * KK * 3 + 0];
    bnd[((size_t)b * NN + h0) * DD + d] = query[b * DD + d];
}

// ---------------- relation-graph scatter (tiny: 2048 edges, hw f32 atomics) --------

__global__ void rel_scatter(const float* __restrict__ x, const float* __restrict__ emb,
                            const int* __restrict__ rsrc, const int* __restrict__ rdst,
                            const int* __restrict__ rtyp, float* __restrict__ agg) {
    int idx = blockIdx.x * blockDim.x + threadIdx.x;
    if (idx >= BB * ERN * 16) return;
    int q = idx & 15;
    int e = (idx >> 4) % ERN;
    int b = idx / (16 * ERN);
    int s = rsrc[e], d = rdst[e], t = rtyp[e];
    float4 xv = ((const float4*)x)[(b * RR + s) * 16 + q];
    float4 ev = ((const float4*)emb)[t * 16 + q];
    float* dp = agg + (b * RR + d) * DD + q * 4;
    unsafeAtomicAdd(dp + 0, xv.x * ev.x);
    unsafeAtomicAdd(dp + 1, xv.y * ev.y);
    unsafeAtomicAdd(dp + 2, xv.z * ev.z);
    unsafeAtomicAdd(dp + 3, xv.w * ev.w);
}

// ---------------- CSR build for the entity graph (once per call) ----------------

__global__ void deg_count(const int* __restrict__ dst, int* __restrict__ deg) {
    int e = blockIdx.x * blockDim.x + threadIdx.x;
    if (e < EE) atomicAdd(&deg[dst[e]], 1);
}

// single-block exclusive scan: rowptr[0]=0, rowptr[i+1]=sum(deg[0..i])
__global__ void scan_block(const int* __restrict__ deg, int* __restrict__ rowptr, int n) {
    __shared__ int sh[256];
    __shared__ int carry;
    int tid = threadIdx.x;
    if (tid == 0) { carry = 0; rowptr[0] = 0; }
    for (int base = 0; base < n; base += 256) {
        int v = (base + tid < n) ? deg[base + tid] : 0;
        __syncthreads();
        sh[tid] = v;
        __syncthreads();
        for (int off = 1; off < 256; off <<= 1) {
            int a = (tid >= off) ? sh[tid - off] : 0;
            __syncthreads();
            sh[tid] += a;
            __syncthreads();
        }
        int c = carry;
        if (base + tid < n) rowptr[base + tid + 1] = c + sh[tid];
        __syncthreads();
        if (tid == 0) carry = c + sh[255];
    }
}

__global__ void csr_fill(const int* __restrict__ src, const int* __restrict__ dst,
                         const int* __restrict__ etyp, int* __restrict__ cursor,
                         int* __restrict__ esrc_s, int* __restrict__ etyp_s) {
    int e = blockIdx.x * blockDim.x + threadIdx.x;
    if (e >= EE) return;
    int slot = atomicAdd(&cursor[dst[e]], 1);
    esrc_s[slot] = src[e];
    etyp_s[slot] = etyp[e];
}

// ---------------- entity aggregation: gather-based, atomic-free ----------------
// agg[b,n,:] = boundary[b,n,:] + sum_{e: dst==n} x[b,src[e],:] * rel[b,etype[e],:]
__global__ void ent_aggregate(const float* __restrict__ x, const float* __restrict__ rel,
                              const float* __restrict__ bnd, const int* __restrict__ rowptr,
                              const int* __restrict__ esrc, const int* __restrict__ etyp,
                              float* __restrict__ agg) {
    int idx = blockIdx.x * blockDim.x + threadIdx.x;
    if (idx >= BB * NN * 16) return;
    int q = idx & 15;
    int n = (idx >> 4) % NN;
    int b = idx / (16 * NN);
    const float4* x4 = (const float4*)x;
    const float4* r4 = (const float4*)rel;
    float4 acc = ((const float4*)bnd)[((size_t)b * NN + n) * 16 + q];
    int je = rowptr[n + 1];
    for (int j = rowptr[n]; j < je; ++j) {
        int s = esrc[j], t = etyp[j];
        float4 xv = x4[((size_t)b * NN + s) * 16 + q];
        float4 rv = r4[(b * RR + t) * 16 + q];
        acc.x += xv.x * rv.x;
        acc.y += xv.y * rv.y;
        acc.z += xv.z * rv.z;
        acc.w += xv.w * rv.w;
    }
    ((float4*)agg)[((size_t)b * NN + n) * 16 + q] = acc;
}

// ---------------- fused WMMA conv linear + LN + relu + residual ----------------
// out (== agg buffer, in-place per 16-row tile):
//   out[i,:] = relu(LN(concat(x[i,:], agg[i,:]) @ W + bias)) + x[i,:]
__global__ void __launch_bounds__(128)
conv_wmma(const float* __restrict__ x, float* __restrict__ agg_out,
          const float* __restrict__ W, const float* __restrict__ bias,
          const float* __restrict__ g, const float* __restrict__ beta, int mtiles) {
    __shared__ float Ws[128][66];
    __shared__ float As[16][68];
    __shared__ float Ct[16][66];

    const int tid = threadIdx.x;
    const int wave = tid >> 5, lane = tid & 31;
    const int half = lane >> 4, m = lane & 15;
    const int colBase = wave * 16;
    const int col = colBase + m;

    for (int i = tid; i < 2048; i += 128) {
        float4 v = ((const float4*)W)[i];
        int r = i >> 4, c = (i & 15) * 4;
        Ws[r][c] = v.x; Ws[r][c + 1] = v.y; Ws[r][c + 2] = v.z; Ws[r][c + 3] = v.w;
    }

    for (int tile = blockIdx.x; tile < mtiles; tile += gridDim.x) {
        __syncthreads();
        const size_t rowBase = (size_t)tile * 16;
        for (int i = tid; i < 512; i += 128) {
            int r = i >> 5, c4 = i & 31;
            float4 v = (c4 < 16)
                ? ((const float4*)x)[(rowBase + r) * 16 + c4]
                : ((const float4*)agg_out)[(rowBase + r) * 16 + (c4 - 16)];
            int c = c4 * 4;
            As[r][c] = v.x; As[r][c + 1] = v.y; As[r][c + 2] = v.z; As[r][c + 3] = v.w;
        }
        __syncthreads();

        v8f c = {};
#pragma unroll
        for (int kk = 0; kk < 32; ++kk) {
            int k0 = kk * 4 + half * 2;
            v2f a, b;
            a.x = As[m][k0]; a.y = As[m][k0 + 1];
            b.x = Ws[k0][col]; b.y = Ws[k0 + 1][col];
            c = __builtin_amdgcn_wmma_f32_16x16x4_f32(false, a, false, b,
                                                      (short)0, c, false, false);
        }
#pragma unroll
        for (int i = 0; i < 8; ++i)
            Ct[half * 8 + i][col] = c[i] + bias[col];
        __syncthreads();

        int row = tid >> 3, cg = tid & 7;
        float vals[8], s = 0.f, s2 = 0.f;
#pragma unroll
        for (int j = 0; j < 8; ++j) {
            float z = Ct[row][cg * 8 + j];
            vals[j] = z; s += z; s2 += z * z;
        }
        s += __shfl_xor(s, 1);  s += __shfl_xor(s, 2);  s += __shfl_xor(s, 4);
        s2 += __shfl_xor(s2, 1); s2 += __shfl_xor(s2, 2); s2 += __shfl_xor(s2, 4);
        float mean = s * (1.f / 64.f);
        float var = s2 * (1.f / 64.f) - mean * mean;
        float inv = rsqrtf(var + 1e-5f);
        size_t gr = rowBase + row;
#pragma unroll
        for (int j = 0; j < 8; ++j) {
            int cc = cg * 8 + j;
            float nrm = (vals[j] - mean) * inv * g[cc] + beta[cc];
            agg_out[gr * DD + cc] = fmaxf(nrm, 0.f) + x[gr * DD + cc];
        }
    }
}

// ---------------- WMMA projection: rel = relu(X@W1+b1)@W2+b2, X=[256,64] ----------------
__global__ void __launch_bounds__(128)
proj_wmma(const float* __restrict__ X, const float* __restrict__ W1,
          const float* __restrict__ b1, const float* __restrict__ W2,
          const float* __restrict__ b2, float* __restrict__ out) {
    __shared__ float W1s[64][66];
    __shared__ float W2s[64][66];
    __shared__ float Xs[16][68];
    __shared__ float Hs[16][68];

    const int tid = threadIdx.x;
    const int wave = tid >> 5, lane = tid & 31;
    const int half = lane >> 4, m = lane & 15;
    const int col = wave * 16 + m;
    const int rowBase = blockIdx.x * 16;

    for (int i = tid; i < 1024; i += 128) {
        float4 v1 = ((const float4*)W1)[i];
        float4 v2 = ((const float4*)W2)[i];
        int r = i >> 4, c = (i & 15) * 4;
        W1s[r][c] = v1.x; W1s[r][c + 1] = v1.y; W1s[r][c + 2] = v1.z; W1s[r][c + 3] = v1.w;
        W2s[r][c] = v2.x; W2s[r][c + 1] = v2.y; W2s[r][c + 2] = v2.z; W2s[r][c + 3] = v2.w;
    }
    for (int i = tid; i < 256; i += 128) {
        float4 v = ((const float4*)X)[rowBase * 16 + i];
        int r = i >> 4, c = (i & 15) * 4;
        Xs[r][c] = v.x; Xs[r][c + 1] = v.y; Xs[r][c + 2] = v.z; Xs[r][c + 3] = v.w;
    }
    __syncthreads();

    v8f h = {};
#pragma unroll
    for (int kk = 0; kk < 16; ++kk) {
        int k0 = kk * 4 + half * 2;
        v2f a, b;
        a.x = Xs[m][k0]; a.y = Xs[m][k0 + 1];
        b.x = W1s[k0][col]; b.y = W1s[k0 + 1][col];
        h = __builtin_amdgcn_wmma_f32_16x16x4_f32(false, a, false, b,
                                                  (short)0, h, false, false);
    }
#pragma unroll
    for (int i = 0; i < 8; ++i)
        Hs[half * 8 + i][col] = fmaxf(h[i] + b1[col], 0.f);
    __syncthreads();

    v8f o = {};
#pragma unroll
    for (int kk = 0; kk < 16; ++kk) {
        int k0 = kk * 4 + half * 2;
        v2f a, b;
        a.x = Hs[m][k0]; a.y = Hs[m][k0 + 1];
        b.x = W2s[k0][col]; b.y = W2s[k0 + 1][col];
        o = __builtin_amdgcn_wmma_f32_16x16x4_f32(false, a, false, b,
                                                  (short)0, o, false, false);
    }
#pragma unroll
    for (int i = 0; i < 8; ++i)
        out[(rowBase + half * 8 + i) * DD + col] = o[i] + b2[col];
}

// ---------------- final MLP: score[b,k] ----------------
__global__ void __launch_bounds__(128)
mlp_kernel(const float* __restrict__ x, const float* __restrict__ query,
           const int* __restrict__ triples, const float* __restrict__ W1,
           const float* __restrict__ b1, const float* __restrict__ W2,
           const float* __restrict__ b2, float* __restrict__ out) {
    int bk = blockIdx.x;
    int b = bk / KK, k = bk % KK;
    int tid = threadIdx.x;
    __shared__ float feat[128];
    __shared__ float red[128];
    int t = triples[(b * KK + k) * 3 + 1];
    feat[tid] = (tid < 64) ? x[((size_t)b * NN + t) * DD + tid]
                           : query[b * DD + (tid - 64)];
    __syncthreads();
    float h = b1[tid];
    for (int i = 0; i < 128; ++i) h += feat[i] * W1[i * 128 + tid];
    h = fmaxf(h, 0.f);
    red[tid] = h * W2[tid];
    __syncthreads();
    for (int s = 64; s > 0; s >>= 1) {
        if (tid < s) red[tid] += red[tid + s];
        __syncthreads();
    }
    if (tid == 0) out[bk] = red[0] + b2[0];
}

// ---------------- host ----------------

extern "C" void kernel_launch(void* const* d_in, const int* in_sizes, int n_in,
                              void* d_out, int out_size, void* d_ws, size_t ws_size,
                              hipStream_t stream) {
    const int* edge_index     = (const int*)d_in[0];   // (2,E)
    const int* edge_type      = (const int*)d_in[1];   // (E,)
    const int* rel_edge_index = (const int*)d_in[2];   // (2,ER)
    const int* rel_edge_type  = (const int*)d_in[3];   // (ER,)
    const int* batch_triples  = (const int*)d_in[4];   // (B,K,3)
    const float* rel_emb  = (const float*)d_in[5];     // (L,RT,D)
    const float* rel_W    = (const float*)d_in[6];     // (L,128,64)
    const float* rel_b    = (const float*)d_in[7];
    const float* rel_g    = (const float*)d_in[8];
    const float* rel_beta = (const float*)d_in[9];
    const float* proj_W1  = (const float*)d_in[10];    // (L,64,64)
    const float* proj_b1  = (const float*)d_in[11];
    const float* proj_W2  = (const float*)d_in[12];
    const float* proj_b2  = (const float*)d_in[13];
    const float* ent_W    = (const float*)d_in[14];    // (L,128,64)
    const float* ent_b    = (const float*)d_in[15];
    const float* ent_g    = (const float*)d_in[16];
    const float* ent_beta = (const float*)d_in[17];
    const float* mlp_W1   = (const float*)d_in[18];    // (128,128)
    const float* mlp_b1   = (const float*)d_in[19];
    const float* mlp_W2   = (const float*)d_in[20];    // (128,1)
    const float* mlp_b2   = (const float*)d_in[21];
    float* out = (float*)d_out;

    const size_t BRD = (size_t)BB * RR * DD;       // 16384
    const size_t BND = (size_t)BB * NN * DD;       // 12,800,000
    float* ws = (float*)d_ws;
    size_t off = 0;
    auto alloc = [&](size_t n) { float* p = ws + off; off += (n + 63) & ~(size_t)63; return p; };
    float* relX    = alloc(BRD);
    float* relAgg  = alloc(BRD);
    float* relBnd  = alloc(BRD);
    float* relProj = alloc(BRD);
    float* query   = alloc(BB * DD);
    float* entBnd  = alloc(BND);
    float* entA    = alloc(BND);
    float* entB    = alloc(BND);
    int* degcur  = (int*)alloc(NN);          // degree, then fill-cursor
    int* rowptr  = (int*)alloc(NN + 64);
    int* esrc_s  = (int*)alloc(EE);
    int* etyp_s  = (int*)alloc(EE);
    (void)ws_size; (void)in_sizes; (void)n_in; (void)out_size;

    const int* e_src = edge_index;
    const int* e_dst = edge_index + EE;

    // ---- build CSR of entity graph by destination (once per call) ----
    zero_f4<<<64, 256, 0, stream>>>((float4*)degcur, NN / 4);
    deg_count<<<(EE + 255) / 256, 256, 0, stream>>>(e_dst, degcur);
    scan_block<<<1, 256, 0, stream>>>(degcur, rowptr, NN);
    copy_f4<<<64, 256, 0, stream>>>((float4*)degcur, (const float4*)rowptr, NN / 4);
    csr_fill<<<(EE + 255) / 256, 256, 0, stream>>>(e_src, e_dst, edge_type,
                                                   degcur, esrc_s, etyp_s);

    // ---- relation graph conv ----
    zero_f4<<<16, 256, 0, stream>>>((float4*)relBnd, (int)(BRD / 4));
    set_rel_boundary<<<1, 256, 0, stream>>>(relBnd, batch_triples);
    copy_f4<<<16, 256, 0, stream>>>((float4*)relX, (const float4*)relBnd, (int)(BRD / 4));
    for (int l = 0; l < NLAY; ++l) {
        copy_f4<<<16, 256, 0, stream>>>((float4*)relAgg, (const float4*)relBnd, (int)(BRD / 4));
        rel_scatter<<<(BB * ERN * 16) / 256, 256, 0, stream>>>(
            relX, rel_emb + (size_t)l * RTN * DD,
            rel_edge_index, rel_edge_index + ERN, rel_edge_type, relAgg);
        conv_wmma<<<16, 128, 0, stream>>>(
            relX, relAgg, rel_W + (size_t)l * 128 * 64,
            rel_b + l * DD, rel_g + l * DD, rel_beta + l * DD, (BB * RR) / 16);
        float* tmp = relX; relX = relAgg; relAgg = tmp;
    }

    // ---- entity graph conv ----
    extract_query<<<1, 256, 0, stream>>>(relX, batch_triples, query);
    zero_f4<<<4096, 256, 0, stream>>>((float4*)entBnd, (int)(BND / 4));
    set_ent_boundary<<<1, 256, 0, stream>>>(entBnd, query, batch_triples);
    copy_f4<<<4096, 256, 0, stream>>>((float4*)entA, (const float4*)entBnd, (int)(BND / 4));
    for (int l = 0; l < NLAY; ++l) {
        proj_wmma<<<16, 128, 0, stream>>>(
            relX, proj_W1 + (size_t)l * DD * DD, proj_b1 + l * DD,
            proj_W2 + (size_t)l * DD * DD, proj_b2 + l * DD, relProj);
        ent_aggregate<<<(BB * NN * 16) / 256, 256, 0, stream>>>(
            entA, relProj, entBnd, rowptr, esrc_s, etyp_s, entB);
        conv_wmma<<<2048, 128, 0, stream>>>(
            entA, entB, ent_W + (size_t)l * 128 * 64,
            ent_b + l * DD, ent_g + l * DD, ent_beta + l * DD, (BB * NN) / 16);
        float* tmp = entA; entA = entB; entB = tmp;
    }

    // ---- final MLP ----
    mlp_kernel<<<BB * KK, 128, 0, stream>>>(entA, query, batch_triples,
                                            mlp_W1, mlp_b1, mlp_W2, mlp_b2, out);
}